// InplaceUpsamplerCUDA_69140383531618
// MI455X (gfx1250) — compile-verified
//
#include <hip/hip_runtime.h>
#include <stdint.h>

// ---------------------------------------------------------------------------
// pixel_shuffle(2) + depthwise 4x4 FIR (taps [1,3,3,1]^2 / 64), pad 2.
// In:  x [16,128,128,128] f32   Out: [16,32,257,257] f32
// Memory-bound. One wave32 per 16x16 output tile. Separable filter as banded
// 16x16 matmuls on WMMA (V_WMMA_F32_16X16X4_F32); input staged via async
// global->LDS loads. Only the live 19x19 filter footprint is fetched; the
// rest of the 32x32 WMMA operand slab is zero-filled in LDS (the band
// matrices are structurally zero there).
// ---------------------------------------------------------------------------

typedef __attribute__((ext_vector_type(2))) float v2f;
typedef __attribute__((ext_vector_type(8))) float v8f;

#define TILES 17      // ceil(257/16)
#define OUTHW 257
#define CIN   128
#define COUT  32
#define HIN   128
#define NB    16

// taps/8: separable factor of outer([1,3,3,1],[1,3,3,1])/64 (symmetric: flip = id)
__device__ __forceinline__ float ftap(int j) {
  if (j == 1 || j == 2) return 0.375f;
  if (j == 0 || j == 3) return 0.125f;
  return 0.0f;
}

__device__ __forceinline__ v8f wmma4(v2f a, v2f b, v8f c) {
#if __has_builtin(__builtin_amdgcn_wmma_f32_16x16x4_f32)
  // 8 args: (neg_a, A, neg_b, B, c_mod, C, reuse_a, reuse_b)
  return __builtin_amdgcn_wmma_f32_16x16x4_f32(false, a, false, b, (short)0, c,
                                               false, false);
#else
  return c;
#endif
}

__device__ __forceinline__ void async_ld_b32(const float* g, float* l) {
#if __has_builtin(__builtin_amdgcn_global_load_async_to_lds_b32)
  __builtin_amdgcn_global_load_async_to_lds_b32(
      (__attribute__((address_space(1))) int*)g,
      (__attribute__((address_space(3))) int*)l, 0, 0);
#else
  *l = *g;  // synchronous fallback
#endif
}

__device__ __forceinline__ void wait_async0() {
#if __has_builtin(__builtin_amdgcn_s_wait_asynccnt)
  __builtin_amdgcn_s_wait_asynccnt(0);
#else
  asm volatile("s_wait_asynccnt 0" ::: "memory");
#endif
  asm volatile("" ::: "memory");
}

__device__ __forceinline__ void wait_ds0() {
#if __has_builtin(__builtin_amdgcn_s_wait_dscnt)
  __builtin_amdgcn_s_wait_dscnt(0);
#else
  asm volatile("s_wait_dscnt 0" ::: "memory");
#endif
  asm volatile("" ::: "memory");
}

__global__ __launch_bounds__(256) void InplaceUpsamplerCUDA_69140383531618_kernel(
    const float* __restrict__ xin, float* __restrict__ out) {
  // Per-wave LDS: 32x32 operand slab (stride 33) + 32x16 intermediate (stride 17)
  __shared__ float P[8][32 * 33];
  __shared__ float T[8][32 * 17];

  const int w    = threadIdx.x >> 5;
  const int lane = threadIdx.x & 31;
  const int half = lane >> 4;   // lane half selects K pair {0,1} vs {2,3} in A/B layouts
  const int l16  = lane & 15;

  const int wid = blockIdx.x * 8 + w;      // grid sized exactly: no stragglers
  const int tx = wid % TILES;
  const int ty = (wid / TILES) % TILES;
  const int c  = (wid / (TILES * TILES)) % COUT;
  const int b  =  wid / (TILES * TILES * COUT);

  const int y0 = ty * 16, x0 = tx * 16;
  float* Pw = P[w];
  float* Tw = T[w];

  // ---- Zero the operand slab (WMMA reads all 32x32; only 19x19 is live data)
  #pragma unroll
  for (int i = lane; i < 32 * 33; i += 32) Pw[i] = 0.0f;
  wait_ds0();  // zeros committed before async LDS writes can land

  // ---- Stage the live 19x19 pixel-shuffled footprint
  //      (X rows y0-2..y0+16, cols x0-2..x0+16), zero-padded by the slab.
  {
    const int  sv   = x0 - 2 + lane;
    const bool vcol = (lane < 19) && (((unsigned)sv) < 256u);
    const int  chb  = c << 2;
    for (int u = 0; u < 19; ++u) {
      const int su = y0 - 2 + u;
      if (vcol && ((unsigned)su) < 256u) {
        // pixel_shuffle gather: X[su][sv] = x[b, 4c + 2*(su&1) + (sv&1), su>>1, sv>>1]
        const int ch = chb + ((su & 1) << 1) + (sv & 1);
        const float* g = xin +
            ((((size_t)b * CIN + ch) * HIN + (su >> 1)) * HIN + (sv >> 1));
        async_ld_b32(g, &Pw[u * 33 + lane]);
      }
    }
  }
  wait_async0();

  // ---- Per-lane band-matrix constants. The A-operand value of the vertical
  //      band G and the B-operand value of the horizontal band F coincide.
  float b0[4][2], b1[4][2];
  #pragma unroll
  for (int k = 0; k < 4; ++k) {
    const int kk = 4 * k + 2 * half;
    b0[k][0] = ftap(kk - l16);        // band(K - idx), K in {kk, kk+1}
    b0[k][1] = ftap(kk + 1 - l16);
    b1[k][0] = ftap(kk + 16 - l16);   // shifted band for the second 16-block
    b1[k][1] = ftap(kk + 17 - l16);
  }

  // ---- Horizontal pass: T(32x16) = [In0 In1](32x32) x [F0;F1], via WMMA.
  #pragma unroll
  for (int rb = 0; rb < 2; ++rb) {
    const int row = rb * 16 + l16;     // A layout: M = l16 in both lane halves
    v8f t = {0.f, 0.f, 0.f, 0.f, 0.f, 0.f, 0.f, 0.f};
    #pragma unroll
    for (int k = 0; k < 4; ++k) {      // In block cols 0..15  x F0
      const int col = 4 * k + 2 * half;
      v2f a; a.x = Pw[row * 33 + col]; a.y = Pw[row * 33 + col + 1];
      v2f f; f.x = b0[k][0];           f.y = b0[k][1];
      t = wmma4(a, f, t);
    }
    #pragma unroll
    for (int k = 0; k < 4; ++k) {      // In block cols 16..31 x F1 (cols 16-18 live)
      const int col = 16 + 4 * k + 2 * half;
      v2f a; a.x = Pw[row * 33 + col]; a.y = Pw[row * 33 + col + 1];
      v2f f; f.x = b1[k][0];           f.y = b1[k][1];
      t = wmma4(a, f, t);
    }
    // D layout -> LDS (rows g and g+8 per VGPR), stride 17 dodges conflicts
    #pragma unroll
    for (int g = 0; g < 8; ++g) {
      Tw[(rb * 16 + g + 8 * half) * 17 + l16] = t[g];
    }
  }
  wait_ds0();  // T visible to cross-lane B-layout reads (same wave, in-order DS)

  // ---- Vertical pass: Out(16x16) = G0 x T0 + G1 x T1
  v8f o = {0.f, 0.f, 0.f, 0.f, 0.f, 0.f, 0.f, 0.f};
  #pragma unroll
  for (int k = 0; k < 4; ++k) {
    const int tr = 4 * k + 2 * half;   // B layout: K pair (+0/+2) by lane half
    v2f a;  a.x  = b0[k][0];               a.y  = b0[k][1];
    v2f bq; bq.x = Tw[tr * 17 + l16];      bq.y = Tw[(tr + 1) * 17 + l16];
    o = wmma4(a, bq, o);
  }
  #pragma unroll
  for (int k = 0; k < 4; ++k) {
    const int tr = 16 + 4 * k + 2 * half;  // T rows 16-18 live, rest are zeros
    v2f a;  a.x  = b1[k][0];               a.y  = b1[k][1];
    v2f bq; bq.x = Tw[tr * 17 + l16];      bq.y = Tw[(tr + 1) * 17 + l16];
    o = wmma4(a, bq, o);
  }

  // ---- Store 16x16 tile (D layout), masked to the 257x257 plane, streaming.
  const size_t obase = ((size_t)(b * COUT + c)) * OUTHW * OUTHW;
  #pragma unroll
  for (int g = 0; g < 8; ++g) {
    const int y = y0 + g + 8 * half;
    const int x = x0 + l16;
    if (y < OUTHW && x < OUTHW)
      __builtin_nontemporal_store(o[g], &out[obase + (size_t)y * OUTHW + x]);
  }
}

extern "C" void kernel_launch(void* const* d_in, const int* in_sizes, int n_in,
                              void* d_out, int out_size, void* d_ws, size_t ws_size,
                              hipStream_t stream) {
  (void)in_sizes; (void)n_in; (void)out_size; (void)d_ws; (void)ws_size;
  const float* x = (const float*)d_in[0];
  // d_in[1] is the fixed normalized [1,3,3,1] outer-product kernel; its exact
  // separable taps (0.125/0.375) are baked into the band matrices above.
  float* out = (float*)d_out;
  const int total_tiles = NB * COUT * TILES * TILES;   // 147,968
  dim3 grid(total_tiles / 8), block(256);              // 8 waves/block
  hipLaunchKernelGGL(InplaceUpsamplerCUDA_69140383531618_kernel,
                     grid, block, 0, stream, x, out);
}